// SpatialLikelihoodGrid_20074677141777
// MI455X (gfx1250) — compile-verified
//
#include <hip/hip_runtime.h>
#include <math.h>

#define NFFT   32768
#define NMASK  32767
#define TPB    512
#define NGRID  200

typedef __attribute__((ext_vector_type(2))) float v2f;
typedef __attribute__((ext_vector_type(8))) float v8f;
typedef int v4i_vs __attribute__((vector_size(16)));          // matches builtin param
typedef __attribute__((address_space(3))) v4i_vs* lds_v4i_p;  // LDS-side pointer

#if defined(__has_builtin)
#if __has_builtin(__builtin_amdgcn_global_load_async_to_lds_b128)
#define HAVE_ASYNC_LDS 1
#endif
#endif

// D(16x16 f32) = A(16x4 f32) * B(4x16 f32) + C  -- chained over K in steps of 4.
static __device__ __forceinline__ v8f wmma4(v2f a, v2f b, v8f c) {
  return __builtin_amdgcn_wmma_f32_16x16x4_f32(false, a, false, b, (short)0, c, false, false);
}

// Hardware V_SIN_F32/V_COS_F32: input in revolutions (D = sin(2*pi*S0)).
// Our twiddle angles are exact revolution fractions -> no range reduction needed.
static __device__ __forceinline__ void sincos_rev(float rev, float& s, float& c) {
  s = __builtin_amdgcn_sinf(rev);
  c = __builtin_amdgcn_cosf(rev);
}

static constexpr float COS16[16] = {
   1.0f,  0.9238795325112867f,  0.7071067811865476f,  0.3826834323650898f,
   0.0f, -0.3826834323650898f, -0.7071067811865476f, -0.9238795325112867f,
  -1.0f, -0.9238795325112867f, -0.7071067811865476f, -0.3826834323650898f,
   0.0f,  0.3826834323650898f,  0.7071067811865476f,  0.9238795325112867f };
static constexpr float SIN16[16] = {
   0.0f,  0.3826834323650898f,  0.7071067811865476f,  0.9238795325112867f,
   1.0f,  0.9238795325112867f,  0.7071067811865476f,  0.3826834323650898f,
   0.0f, -0.3826834323650898f, -0.7071067811865476f, -0.9238795325112867f,
  -1.0f, -0.9238795325112867f, -0.7071067811865476f, -0.3826834323650898f };
static constexpr float COS8[8] = {
   1.0f,  0.7071067811865476f,  0.0f, -0.7071067811865476f,
  -1.0f, -0.7071067811865476f,  0.0f,  0.7071067811865476f };
static constexpr float SIN8[8] = {
   0.0f,  0.7071067811865476f,  1.0f,  0.7071067811865476f,
   0.0f, -0.7071067811865476f, -1.0f, -0.7071067811865476f };

// One radix-16 stage with span S over 32768 complex points in LDS.
// Forward (INV=false): DIF, twiddle e^{-2pi i j r/(16S)} applied to outputs.
// Inverse (INV=true):  DIT, twiddle e^{+2pi i j q/(16S)} applied to inputs.
// 2048 butterflies/stage; 16 butterflies per WMMA block; 128 blocks over 16 waves.
template<int S, bool INV>
static __device__ __forceinline__ void radix16_stage(float* __restrict__ re,
                                                     float* __restrict__ im) {
  const int tid  = threadIdx.x;
  const int lane = tid & 31;
  const int wave = tid >> 5;
  const int half = lane >> 4;   // 0: rows k, k+1 ; 1: rows k+2, k+3 (A/B layout)
  const int col  = lane & 15;

  // A fragments: DFT16 matrix. Row m = col; chunk t holds K = 4t .. 4t+3.
  v2f Ar[4], Ai[4];
#pragma unroll
  for (int t = 0; t < 4; ++t) {
    const int k0 = 4 * t + 2 * half;
    const int e0 = (col * k0) & 15;
    const int e1 = (col * (k0 + 1)) & 15;
    v2f ar; ar[0] = COS16[e0]; ar[1] = COS16[e1];
    v2f ai;
    if (INV) { ai[0] =  SIN16[e0]; ai[1] =  SIN16[e1]; }
    else     { ai[0] = -SIN16[e0]; ai[1] = -SIN16[e1]; }
    Ar[t] = ar; Ai[t] = ai;
  }

  const float rev16S = 1.0f / (16.0f * (float)S);  // revolutions per unit j*q

#pragma unroll 1
  for (int bi = 0; bi < 8; ++bi) {
    const int u    = (wave * 8 + bi) * 16 + col;  // this lane's butterfly (column)
    const int g    = u / S;                       // group
    const int j    = u - g * S;                   // offset within group
    const int base = g * (16 * S) + j;            // element q lives at base + q*S

    // B fragments: X[q][col], rows chunked by 4 like A.
    v2f Xr[4], Xi[4];
#pragma unroll
    for (int t = 0; t < 4; ++t) {
      const int q0 = 4 * t + 2 * half;
      float xr0 = re[base + q0 * S],       xi0 = im[base + q0 * S];
      float xr1 = re[base + (q0 + 1) * S], xi1 = im[base + (q0 + 1) * S];
      if (INV) {
        float s, c;
        sincos_rev((float)(j * q0) * rev16S, s, c);
        float tr = xr0 * c - xi0 * s; xi0 = xr0 * s + xi0 * c; xr0 = tr;
        sincos_rev((float)(j * (q0 + 1)) * rev16S, s, c);
        tr = xr1 * c - xi1 * s; xi1 = xr1 * s + xi1 * c; xr1 = tr;
      }
      v2f vr; vr[0] = xr0; vr[1] = xr1;
      v2f vi; vi[0] = xi0; vi[1] = xi1;
      Xr[t] = vr; Xi[t] = vi;
    }

    // Complex 16x16 matmul: Ur = Wr*Xr - Wi*Xi ; Ui = Wr*Xi + Wi*Xr
    v8f ur = {0.f,0.f,0.f,0.f,0.f,0.f,0.f,0.f};
    v8f t2 = {0.f,0.f,0.f,0.f,0.f,0.f,0.f,0.f};
    v8f ui = {0.f,0.f,0.f,0.f,0.f,0.f,0.f,0.f};
#pragma unroll
    for (int t = 0; t < 4; ++t) ur = wmma4(Ar[t], Xr[t], ur);
#pragma unroll
    for (int t = 0; t < 4; ++t) t2 = wmma4(Ai[t], Xi[t], t2);
#pragma unroll
    for (int t = 0; t < 4; ++t) ui = wmma4(Ar[t], Xi[t], ui);
#pragma unroll
    for (int t = 0; t < 4; ++t) ui = wmma4(Ai[t], Xr[t], ui);
    ur = ur - t2;

    // D layout: v holds row (v + 8*half), column = col. Scatter back in place.
#pragma unroll
    for (int v = 0; v < 8; ++v) {
      const int r = v + 8 * half;
      float yr = ur[v], yi = ui[v];
      if (!INV) {
        float s, c;
        sincos_rev(-(float)(j * r) * rev16S, s, c);
        float tr = yr * c - yi * s; yi = yr * s + yi * c; yr = tr;
      }
      re[base + r * S] = yr;
      im[base + r * S] = yi;
    }
  }
}

// Span-1 radix-8 stage: plain 8-point DFT on contiguous 8-blocks (no twiddle).
template<bool INV>
static __device__ __forceinline__ void radix8_stage(float* __restrict__ re,
                                                    float* __restrict__ im) {
  const int tid = threadIdx.x;
#pragma unroll 1
  for (int b = tid; b < NFFT / 8; b += TPB) {
    const int base = b * 8;
    float xr[8], xi[8];
#pragma unroll
    for (int q = 0; q < 8; ++q) { xr[q] = re[base + q]; xi[q] = im[base + q]; }
    float yr[8], yi[8];
#pragma unroll
    for (int r = 0; r < 8; ++r) {
      float ar = 0.f, ai = 0.f;
#pragma unroll
      for (int q = 0; q < 8; ++q) {
        const int v = (q * r) & 7;
        const float c = COS8[v];
        const float s = INV ? SIN8[v] : -SIN8[v];
        ar += xr[q] * c - xi[q] * s;
        ai += xr[q] * s + xi[q] * c;
      }
      yr[r] = ar; yi[r] = ai;
    }
#pragma unroll
    for (int r = 0; r < 8; ++r) { re[base + r] = yr[r]; im[base + r] = yi[r]; }
  }
}

__global__ __launch_bounds__(TPB, 1)
void srp_phat_kernel(const float* __restrict__ signal,
                     const float* __restrict__ mics,
                     const float* __restrict__ rooms,
                     const int* __restrict__ pii,
                     const int* __restrict__ pjj,
                     float* __restrict__ out,
                     int n_mics) {
  extern __shared__ float smem[];
  float* re = smem;
  float* im = smem + NFFT;

  const int b   = blockIdx.x;
  const int tid = threadIdx.x;
  const int ii  = pii[0];
  const int jj  = pjj[0];

  // Prefetch the small per-batch geometry data used at the very end.
  __builtin_prefetch(&mics[(size_t)b * n_mics * 3], 0, 3);
  __builtin_prefetch(&rooms[(size_t)b * 3], 0, 3);

  // Packed-real trick: z = x_i + i * x_j, one complex FFT for both channels.
  const float* si = signal + ((size_t)b * n_mics + ii) * NFFT;
  const float* sj = signal + ((size_t)b * n_mics + jj) * NFFT;

#if HAVE_ASYNC_LDS
  // CDNA5 async global->LDS DMA (no VGPR staging), tracked by ASYNCcnt.
  for (int n = tid; n < NFFT / 4; n += TPB) {
    __builtin_amdgcn_global_load_async_to_lds_b128(
        (v4i_vs*)(si + 4 * n), (lds_v4i_p)(re + 4 * n), 0, 0);
    __builtin_amdgcn_global_load_async_to_lds_b128(
        (v4i_vs*)(sj + 4 * n), (lds_v4i_p)(im + 4 * n), 0, 0);
  }
#if __has_builtin(__builtin_amdgcn_s_wait_asynccnt)
  __builtin_amdgcn_s_wait_asynccnt(0);
#else
  asm volatile("s_wait_asynccnt 0" ::: "memory");
#endif
#else
  for (int n = tid; n < NFFT / 4; n += TPB) {
    const float4 a = reinterpret_cast<const float4*>(si)[n];
    const float4 c = reinterpret_cast<const float4*>(sj)[n];
    reinterpret_cast<float4*>(re)[n] = a;
    reinterpret_cast<float4*>(im)[n] = c;
  }
#endif
  __syncthreads();

  // Forward DIF: natural in -> digit-block-reversed out.
  radix16_stage<2048, false>(re, im); __syncthreads();
  radix16_stage<128,  false>(re, im); __syncthreads();
  radix16_stage<8,    false>(re, im); __syncthreads();
  radix8_stage<false>(re, im);        __syncthreads();

  // PHAT cross-spectrum, pointwise in rev order. Storage p holds frequency
  // k = b0|(b1<<4)|(b2<<8)|(b3<<12) for p = (b0<<11)|(b1<<7)|(b2<<3)|b3.
  for (int p = tid; p < NFFT; p += TPB) {
    const int k  = ((p >> 11) & 15) | (((p >> 7) & 15) << 4) |
                   (((p >> 3) & 15) << 8) | ((p & 7) << 12);
    const int kp = (NFFT - k) & NMASK;
    const int pp = ((kp & 15) << 11) | (((kp >> 4) & 15) << 7) |
                   (((kp >> 8) & 15) << 3) | ((kp >> 12) & 7);
    if (p > pp) continue;                       // each Hermitian pair owned once
    const float zr = re[p],  zi = im[p];        // Z[k]
    const float wr = re[pp], wi = im[pp];       // Z[N-k]
    const float xir = 0.5f * (zr + wr), xii = 0.5f * (zi - wi);   // Xi[k]
    const float xjr = 0.5f * (zi + wi), xji = -0.5f * (zr - wr);  // Xj[k]
    float gr = xir * xjr + xii * xji;           // G = Xi * conj(Xj)
    float gi = xii * xjr - xir * xji;
    const float mag = sqrtf(gr * gr + gi * gi);
    const float s = 1.0f / ((mag + 1e-8f) * (float)NFFT);  // PHAT + 1/N of irfft
    gr *= s; gi *= s;
    re[p]  = gr; im[p]  = gi;
    re[pp] = gr; im[pp] = (p == pp) ? gi : -gi; // G[N-k] = conj(G[k])
  }
  __syncthreads();

  // Inverse DIT: rev-order in -> natural out; cc = Re(.) lands in re[].
  radix8_stage<true>(re, im);        __syncthreads();
  radix16_stage<8,    true>(re, im); __syncthreads();
  radix16_stage<128,  true>(re, im); __syncthreads();
  radix16_stage<2048, true>(re, im); __syncthreads();

  // Grid of fractional-delay lookups into cc (LDS-resident).
  const size_t mb  = (size_t)b * n_mics * 3;
  const float mix = mics[mb + (size_t)ii * 3 + 0];
  const float miy = mics[mb + (size_t)ii * 3 + 1];
  const float miz = mics[mb + (size_t)ii * 3 + 2];
  const float mjx = mics[mb + (size_t)jj * 3 + 0];
  const float mjy = mics[mb + (size_t)jj * 3 + 1];
  const float mjz = mics[mb + (size_t)jj * 3 + 2];
  const float rx = rooms[(size_t)b * 3 + 0];
  const float ry = rooms[(size_t)b * 3 + 1];
  const float rz = rooms[(size_t)b * 3 + 2];
  const float gz  = 0.5f * rz;
  const float dzi = gz - miz;
  const float dzj = gz - mjz;
  const float kds = 16000.0f / 343.0f;
  float* ob = out + (size_t)b * (NGRID * NGRID);
  for (int idx = tid; idx < NGRID * NGRID; idx += TPB) {
    const int ix = idx / NGRID;
    const int iy = idx - ix * NGRID;
    const float px = rx * (((float)ix + 0.5f) * (1.0f / NGRID));
    const float py = ry * (((float)iy + 0.5f) * (1.0f / NGRID));
    const float dxi = px - mix, dyi = py - miy;
    const float dxj = px - mjx, dyj = py - mjy;
    const float di = sqrtf(dxi * dxi + dyi * dyi + dzi * dzi);
    const float dj = sqrtf(dxj * dxj + dyj * dyj + dzj * dzj);
    const float delay = (di - dj) * kds;
    const float d0 = floorf(delay);
    const float fr = delay - d0;
    const int i0 = ((int)d0) & NMASK;      // == jnp.mod(int32, 32768)
    const int i1 = (i0 + 1) & NMASK;
    const float v0 = re[i0];
    const float v1 = re[i1];
    ob[idx] = v0 + (v1 - v0) * fr;
  }
}

extern "C" void kernel_launch(void* const* d_in, const int* in_sizes, int n_in,
                              void* d_out, int out_size, void* d_ws, size_t ws_size,
                              hipStream_t stream) {
  (void)n_in; (void)out_size; (void)d_ws; (void)ws_size;
  const float* signal = (const float*)d_in[0];
  const float* mics   = (const float*)d_in[1];
  const float* rooms  = (const float*)d_in[2];
  const int*   pii    = (const int*)d_in[3];
  const int*   pjj    = (const int*)d_in[4];
  float*       out    = (float*)d_out;

  const int B = in_sizes[2] / 3;                       // room_dims is (B, 3)
  if (B <= 0) return;
  const int n_mics = in_sizes[1] / (B * 3);            // mic_coordinates (B, n_mics, 3)

  const size_t shmem = (size_t)2 * NFFT * sizeof(float);  // 256 KB: re + im in LDS
  (void)hipFuncSetAttribute(reinterpret_cast<const void*>(srp_phat_kernel),
                            hipFuncAttributeMaxDynamicSharedMemorySize, (int)shmem);
  srp_phat_kernel<<<B, TPB, shmem, stream>>>(signal, mics, rooms, pii, pjj, out, n_mics);
}